// BiMamba_14353780703876
// MI455X (gfx1250) — compile-verified
//
#include <hip/hip_runtime.h>

typedef float v2f __attribute__((ext_vector_type(2)));
typedef float v8f __attribute__((ext_vector_type(8)));

// Problem dims (fixed by the reference)
#define BSZ 2
#define CC  80     // input channels
#define LL  2048   // sequence length
#define HH  256    // hidden
#define EE  512    // d_inner
#define SS  16     // d_state
#define RR  16     // dt_rank
#define MT  4      // M-tiles per wave in the vectorized GEMM

// ---------------------------------------------------------------------------
// Generic-stride f32 GEMM on V_WMMA_F32_16X16X4_F32 (used only for the input
// projection, where A's K-stride is LL).
//   Out[m][n] = act( sum_k A[m][k] * W[n][k] + bias[n] )
// ---------------------------------------------------------------------------
__global__ void gemm_f32_wmma(const float* __restrict__ A, long a_off0, long a_zoff,
                              long a_rs, long a_cs,
                              const float* __restrict__ W, int ldw,
                              const float* __restrict__ bias,
                              float* __restrict__ Cout, long c_off0, long c_zoff,
                              long c_rs, long c_cs,
                              int K, int act)
{
    const int  lane = threadIdx.x & 31;
    const int  m0   = blockIdx.x << 4;
    const int  n0   = blockIdx.y << 4;
    const long z    = blockIdx.z;
    const long abase = a_off0 + z * a_zoff;
    const long cbase = c_off0 + z * c_zoff;

    const int  half = lane >> 4;        // 0: lanes 0-15, 1: lanes 16-31
    const int  l15  = lane & 15;
    const long aRow = abase + (long)(m0 + l15) * a_rs;
    const long bRow = (long)(n0 + l15) * (long)ldw;

    v8f acc = {0.f, 0.f, 0.f, 0.f, 0.f, 0.f, 0.f, 0.f};
    for (int k = 0; k < K; k += 4) {
        // ISA layout (16x4 f32 A): VGPR0 holds K=k (lanes 0-15) / K=k+2 (lanes 16-31),
        // VGPR1 holds K=k+1 / K=k+3. B (4x16) mirrors with N striped across lanes.
        const int kk = k + (half << 1);
        v2f av, bv;
        av.x = A[aRow + (long)kk       * a_cs];
        av.y = A[aRow + (long)(kk + 1) * a_cs];
        bv.x = W[bRow + kk];
        bv.y = W[bRow + kk + 1];
        acc = __builtin_amdgcn_wmma_f32_16x16x4_f32(false, av, false, bv,
                                                    (short)0, acc, false, false);
    }

    const int   n  = n0 + l15;
    const float bb = bias ? bias[n] : 0.f;
#pragma unroll
    for (int v = 0; v < 8; ++v) {
        const int m   = m0 + v + (half << 3);
        float     val = acc[v] + bb;
        if (act == 1) val = (val > 20.f) ? val : log1pf(__expf(val));  // softplus
        Cout[cbase + (long)m * c_rs + (long)n * c_cs] = val;
    }
}

// ---------------------------------------------------------------------------
// Vectorized, M-blocked f32 WMMA GEMM for contiguous-K A (a_cs == 1).
// Each wave computes MT vertical 16x16 tiles sharing one B fragment:
//   - B fragment: one b64 load per 4-K step, reused by MT WMMAs
//   - A fragments: b64 loads (v2f), all addresses 8-byte aligned
// a_rs may be negative (backward-direction reversed view of xp).
// ---------------------------------------------------------------------------
__global__ void gemm_f32_wmma_v(const float* __restrict__ A, long a_off0, long a_zoff,
                                long a_rs,
                                const float* __restrict__ W, int ldw,
                                const float* __restrict__ bias,
                                float* __restrict__ Cout, long c_off0, long c_zoff,
                                long c_rs, long c_cs,
                                int K, int act)
{
    const int  lane = threadIdx.x & 31;
    const int  m0   = blockIdx.x * (16 * MT);
    const int  n0   = blockIdx.y << 4;
    const long z    = blockIdx.z;
    const long abase = a_off0 + z * a_zoff;
    const long cbase = c_off0 + z * c_zoff;

    const int  half = lane >> 4;
    const int  l15  = lane & 15;

    long aRow[MT];
#pragma unroll
    for (int i = 0; i < MT; ++i)
        aRow[i] = abase + (long)(m0 + i * 16 + l15) * a_rs;
    const long bRow = (long)(n0 + l15) * (long)ldw;

    const v8f zacc = {0.f, 0.f, 0.f, 0.f, 0.f, 0.f, 0.f, 0.f};
    v8f acc[MT];
#pragma unroll
    for (int i = 0; i < MT; ++i) acc[i] = zacc;

    for (int k = 0; k < K; k += 4) {
        const int kk = k + (half << 1);                 // even => 8B aligned
        const v2f bv = *(const v2f*)&W[bRow + kk];      // shared by MT WMMAs
#pragma unroll
        for (int i = 0; i < MT; ++i) {
            const v2f av = *(const v2f*)&A[aRow[i] + kk];
            acc[i] = __builtin_amdgcn_wmma_f32_16x16x4_f32(false, av, false, bv,
                                                           (short)0, acc[i], false, false);
        }
    }

    const int   n  = n0 + l15;
    const float bb = bias ? bias[n] : 0.f;
#pragma unroll
    for (int i = 0; i < MT; ++i) {
#pragma unroll
        for (int v = 0; v < 8; ++v) {
            const int m   = m0 + i * 16 + v + (half << 3);
            float     val = acc[i][v] + bb;
            if (act == 1) val = (val > 20.f) ? val : log1pf(__expf(val));
            Cout[cbase + (long)m * c_rs + (long)n * c_cs] = val;
        }
    }
}

// ---------------------------------------------------------------------------
// Depthwise causal conv (width 4) + SiLU over xz's first EE columns.
// ---------------------------------------------------------------------------
__global__ void conv_silu(const float* __restrict__ xz, const float* __restrict__ cw,
                          const float* __restrict__ cb, float* __restrict__ xs)
{
    const long idx = (long)blockIdx.x * blockDim.x + threadIdx.x;  // over NT*EE
    const int  e   = (int)(idx & (EE - 1));
    const long m   = idx >> 9;                 // EE == 512
    const int  l   = (int)(m & (LL - 1));      // LL power of two
    float acc = cb[e];
#pragma unroll
    for (int j = 0; j < 4; ++j) {
        const int ll = l - 3 + j;              // causal: left pad 3
        if (ll >= 0)
            acc = fmaf(cw[e * 4 + j], xz[(m - 3 + j) * (long)(2 * EE) + e], acc);
    }
    xs[m * (long)EE + e] = acc / (1.f + __expf(-acc));  // silu
}

// ---------------------------------------------------------------------------
// Selective scan. One 16-lane group per (b, channel e); lane n owns state n.
//   h = exp(dt*A_n)*h + dt*B_n*xs ;  y_t = sum_n h*C_n  (shfl-xor reduce)
// Epilogue fused: y = (y + xs*D[e]) * silu(z), written to yb.
// ---------------------------------------------------------------------------
__global__ void scan_kernel(const float* __restrict__ dtb, const float* __restrict__ xs,
                            const float* __restrict__ dbl, const float* __restrict__ xz,
                            const float* __restrict__ A_log, const float* __restrict__ Dp,
                            float* __restrict__ yb)
{
    const int gid = blockIdx.x * blockDim.x + threadIdx.x;
    const int n   = gid & 15;
    const int ch  = gid >> 4;             // 0 .. BSZ*EE-1
    const int e   = ch & (EE - 1);
    const int b   = ch >> 9;              // EE == 512

    const float a_n = -__expf(A_log[e * SS + n]);   // A = -exp(A_log)
    const float d_e = Dp[e];
    float h = 0.f;
    const long rowBase = (long)b * LL;

    for (int t = 0; t < LL; ++t) {
        const long  row = rowBase + t;
        const float dt  = dtb[row * EE + e];         // broadcast within group
        const float xv  = xs[row * EE + e];
        const float bm  = dbl[row * 48 + RR + n];        // B_t[n]
        const float cm  = dbl[row * 48 + RR + SS + n];   // C_t[n]
        h = __expf(dt * a_n) * h + dt * bm * xv;
        float p = h * cm;
        p += __shfl_xor(p, 1, 16);
        p += __shfl_xor(p, 2, 16);
        p += __shfl_xor(p, 4, 16);
        p += __shfl_xor(p, 8, 16);
        if (n == 0) {
            float y = p + xv * d_e;
            const float zv = xz[row * (long)(2 * EE) + EE + e];
            y *= zv / (1.f + __expf(-zv));           // * silu(z)
            yb[row * (long)EE + e] = y;
        }
    }
}

// ---------------------------------------------------------------------------
extern "C" void kernel_launch(void* const* d_in, const int* in_sizes, int n_in,
                              void* d_out, int out_size, void* d_ws, size_t ws_size,
                              hipStream_t stream)
{
    (void)in_sizes; (void)n_in; (void)out_size; (void)ws_size;

    const float* x      = (const float*)d_in[0];   // (B, C, L)
    const float* proj_w = (const float*)d_in[1];   // (H, C)
    const float* proj_b = (const float*)d_in[2];   // (H,)

    const long NT = (long)BSZ * LL;                // 4096 rows

    // Workspace layout (floats). Directions run sequentially and reuse buffers.
    float* ws  = (float*)d_ws;
    float* xp  = ws;                    // NT*HH        projected input (shared)
    float* xz  = xp  + NT * HH;         // NT*2*EE      in-proj (xc | z)
    float* xsb = xz  + NT * 2 * EE;     // NT*EE        conv+silu activations
    float* dbl = xsb + NT * EE;         // NT*48        [dt_rank | B | C]
    float* dtb = dbl + NT * 48;         // NT*EE        softplus(dt)
    float* yb  = dtb + NT * EE;         // NT*EE        scan output * silu(z)

    // 1) xp[b,l,h] = sum_c x[b,c,l]*proj_w[h,c] + proj_b[h]   (strided-A GEMM)
    {
        dim3 g(LL / 16, HH / 16, BSZ);
        gemm_f32_wmma<<<g, 32, 0, stream>>>(x, 0L, (long)CC * LL, 1L, (long)LL,
                                            proj_w, CC, proj_b,
                                            xp, 0L, (long)LL * HH, (long)HH, 1L,
                                            CC, 0);
    }

    for (int dir = 0; dir < 2; ++dir) {
        const float* W_in   = (const float*)d_in[3 + dir * 9 + 0];  // (1024, 256)
        const float* conv_w = (const float*)d_in[3 + dir * 9 + 1];  // (512, 4)
        const float* conv_b = (const float*)d_in[3 + dir * 9 + 2];  // (512,)
        const float* W_xp   = (const float*)d_in[3 + dir * 9 + 3];  // (48, 512)
        const float* W_dt   = (const float*)d_in[3 + dir * 9 + 4];  // (512, 16)
        const float* b_dt   = (const float*)d_in[3 + dir * 9 + 5];  // (512,)
        const float* A_log  = (const float*)d_in[3 + dir * 9 + 6];  // (512, 16)
        const float* Dp     = (const float*)d_in[3 + dir * 9 + 7];  // (512,)
        const float* W_out  = (const float*)d_in[3 + dir * 9 + 8];  // (256, 512)

        // 2) xz = xp(dir view) @ W_in^T.  Backward dir: negative row stride = reversal.
        {
            dim3 g(LL / (16 * MT), (2 * EE) / 16, BSZ);
            const long a0 = dir ? (long)(LL - 1) * HH : 0L;
            const long rs = dir ? -(long)HH : (long)HH;
            gemm_f32_wmma_v<<<g, 32, 0, stream>>>(xp, a0, (long)LL * HH, rs,
                                                  W_in, HH, nullptr,
                                                  xz, 0L, (long)LL * 2 * EE, (long)(2 * EE), 1L,
                                                  HH, 0);
        }

        // 3) depthwise conv + silu
        conv_silu<<<(unsigned)((NT * EE) / 256), 256, 0, stream>>>(xz, conv_w, conv_b, xsb);

        // 4) dbl = xs @ W_xproj^T   (N = 48 = dt_rank|B|C)
        {
            dim3 g((unsigned)(NT / (16 * MT)), 48 / 16, 1);
            gemm_f32_wmma_v<<<g, 32, 0, stream>>>(xsb, 0L, 0L, (long)EE,
                                                  W_xp, EE, nullptr,
                                                  dbl, 0L, 0L, 48L, 1L,
                                                  EE, 0);
        }

        // 5) dt = softplus(dbl[:, :16] @ W_dt^T + b_dt)
        {
            dim3 g((unsigned)(NT / (16 * MT)), EE / 16, 1);
            gemm_f32_wmma_v<<<g, 32, 0, stream>>>(dbl, 0L, 0L, 48L,
                                                  W_dt, RR, b_dt,
                                                  dtb, 0L, 0L, (long)EE, 1L,
                                                  RR, 1);
        }

        // 6) selective scan (fused +xs*D and *silu(z))
        scan_kernel<<<(BSZ * EE * 16) / 256, 256, 0, stream>>>(dtb, xsb, dbl, xz,
                                                               A_log, Dp, yb);

        // 7) out[b, dir*256+n, l(m)] = yb @ W_out^T, written transposed into d_out
        {
            dim3 g(LL / (16 * MT), HH / 16, BSZ);
            gemm_f32_wmma_v<<<g, 32, 0, stream>>>(yb, 0L, (long)LL * EE, (long)EE,
                                                  W_out, EE, nullptr,
                                                  (float*)d_out,
                                                  (long)dir * HH * LL,      // feature block
                                                  (long)(2 * HH) * LL,      // per-batch
                                                  1L, (long)LL,             // m->l, n->row
                                                  EE, 0);
        }
    }
}